// SWAP_49941879718225
// MI455X (gfx1250) — compile-verified
//
#include <hip/hip_runtime.h>

// ---------------------------------------------------------------------------
// Problem constants (from reference)
// ---------------------------------------------------------------------------
#define HGT   60
#define WID   108
#define NPIX  (HGT * WID)        // 6480 pixels per (b) image plane
#define NLAY  133
#define NB    4
#define KS    12
#define PH    5
#define PW    9
#define LQ    45                 // number of patches (real columns)
#define LP    48                 // padded to 3 x 16 for WMMA tiles
#define CR    432                // unfold rows = 3 * 12 * 12 (27 x 16 exactly)

typedef __attribute__((ext_vector_type(2))) float v2f;
typedef __attribute__((ext_vector_type(8))) float v8f;

// ---------------------------------------------------------------------------
// Main kernel: one workgroup per (layer, batch). 288 threads = 9 wave32.
//   Stage 1: build Uu (=unfold(I*S*M)) and Ku (=unfold(I*S*(1-M))) in LDS.
//   Stage 2: Wm = Uu^T * Ku        (45x45 padded 48x48, K=432)  [f32 WMMA]
//   Stage 3: L1-normalize rows of Wm
//   Stage 4: WK = Ku * Wm^T        (432x45 padded 48,  K=48)    [f32 WMMA]
//            fold + layer reduction:
//              USE_WS=true : plain stores into per-(b,l) partial planes (ws)
//              USE_WS=false: atomicAdd into the out accumulator
// Dual accumulators in both k-loops break the WMMA RAW dependency chain.
// ---------------------------------------------------------------------------
template <bool USE_WS>
__global__ __launch_bounds__(288) void matting_layer_kernel(
    const float* __restrict__ I,     // (4,3,60,108)
    const float* __restrict__ S,     // (4,60,108,133)
    const float* __restrict__ M,     // (4,1,60,108)
    float* __restrict__ outacc,      // (4,3,60,108) accumulator (atomic path)
    float* __restrict__ part)        // (4,133,3,60,108) partials (ws path)
{
  __shared__ __align__(16) float Uu[CR * LP];   // 82.9 KB
  __shared__ __align__(16) float Ku[CR * LP];   // 82.9 KB
  __shared__ __align__(16) float Wm[LP * LP];   //  9.2 KB   (~175 KB < 320 KB)

  const int l    = blockIdx.x;          // layer
  const int b    = blockIdx.y;          // batch
  const int tid  = threadIdx.x;
  const int lane = tid & 31;
  const int wave = tid >> 5;            // 0..8

  const float* Ib = I + (size_t)b * 3 * NPIX;
  const float* Sb = S + (size_t)b * NPIX * NLAY + l;
  const float* Mb = M + (size_t)b * NPIX;

  // ---- Stage 1a: zero the 3 padding columns (45..47) of Uu/Ku -------------
  for (int e = tid; e < CR * 3; e += 288) {
    int r = e / 3, p = LQ + (e - (e / 3) * 3);
    Uu[r * LP + p] = 0.0f;
    Ku[r * LP + p] = 0.0f;
  }

  // ---- Stage 1b: fill Uu/Ku (unfold is a permutation: 1 write/element) ----
  for (int pix = tid; pix < NPIX; pix += 288) {
    int h  = pix / WID, w = pix - h * WID;
    int ph = h / KS,  k1 = h - ph * KS;
    int pw = w / KS,  k2 = w - pw * KS;
    int p     = ph * PW + pw;           // patch (column) index
    int rbase = k1 * KS + k2;           // within-channel row index
    float s  = Sb[(size_t)pix * NLAY];
    float m  = Mb[pix];
    float sm = s * m;                   // S*M
    float sk = s - sm;                  // S*(1-M)
#pragma unroll
    for (int c = 0; c < 3; ++c) {
      float iv = Ib[c * NPIX + pix];
      int   r  = c * 144 + rbase;
      Uu[r * LP + p] = iv * sm;
      Ku[r * LP + p] = iv * sk;
    }
  }
  __syncthreads();

  // ---- Stage 2: Wm = Uu^T * Ku ; 9 waves <-> 9 (16x16) tiles --------------
  // A(16x4 f32) layout: lane<16: M=lane, VGPRv=K(2h+v) ; lanes>=16: K+=2
  {
    const int ti   = wave / 3, tj = wave - ti * 3;
    const int h    = lane >> 4;
    const int mcol = ti * 16 + (lane & 15);   // output row  (p index)
    const int ncol = tj * 16 + (lane & 15);   // output col  (q index)
    v8f acc0 = {}, acc1 = {};
    for (int k0 = 0; k0 < CR; k0 += 8) {      // 54 iters, 2 indep WMMA chains
      const int ka = k0 + 2 * h;
      v2f a, bb, a2, bb2;
      a.x   = Uu[ka * LP + mcol];
      a.y   = Uu[(ka + 1) * LP + mcol];
      bb.x  = Ku[ka * LP + ncol];
      bb.y  = Ku[(ka + 1) * LP + ncol];
      a2.x  = Uu[(ka + 4) * LP + mcol];
      a2.y  = Uu[(ka + 5) * LP + mcol];
      bb2.x = Ku[(ka + 4) * LP + ncol];
      bb2.y = Ku[(ka + 5) * LP + ncol];
      acc0 = __builtin_amdgcn_wmma_f32_16x16x4_f32(
          false, a, false, bb, (short)0, acc0, false, false);
      acc1 = __builtin_amdgcn_wmma_f32_16x16x4_f32(
          false, a2, false, bb2, (short)0, acc1, false, false);
    }
    // C/D layout: VGPR v -> row v (lanes 0-15) / row v+8 (lanes 16-31)
#pragma unroll
    for (int v = 0; v < 8; ++v) {
      int row = ti * 16 + v + 8 * h;
      Wm[row * LP + ncol] = acc0[v] + acc1[v];
    }
  }
  __syncthreads();

  // ---- Stage 3: L1 row normalization (rows 45..47 are all-zero: 0*1/eps=0)
  if (tid < LQ) {
    float s = 0.0f;
    for (int q = 0; q < LQ; ++q) s += fabsf(Wm[tid * LP + q]);
    float inv = 1.0f / fmaxf(s, 1e-12f);
    for (int q = 0; q < LP; ++q) Wm[tid * LP + q] *= inv;
  }
  __syncthreads();

  // ---- Stage 4: WK = Ku * Wm^T ; 81 tiles over 9 waves, fold + reduce -----
  {
    const int h  = lane >> 4;
    const int lc = lane & 15;
    for (int t = wave; t < 81; t += 9) {          // uniform: 9 iters per wave
      const int mt = t / 3, nt = t - mt * 3;
      const int mrow = mt * 16 + lc;              // unfold row (c,k1,k2)
      const int ncol = nt * 16 + lc;              // patch index i
      v8f acc0 = {}, acc1 = {};
#pragma unroll
      for (int s = 0; s < 6; ++s) {               // 12 k-steps, 2 chains
        const int ka = 8 * s + 2 * h;             // even -> 8B aligned pairs
        v2f a  = *(const v2f*)&Ku[mrow * LP + ka];
        v2f bb = *(const v2f*)&Wm[ncol * LP + ka];
        v2f a2 = *(const v2f*)&Ku[mrow * LP + ka + 4];
        v2f b2 = *(const v2f*)&Wm[ncol * LP + ka + 4];
        acc0 = __builtin_amdgcn_wmma_f32_16x16x4_f32(
            false, a, false, bb, (short)0, acc0, false, false);
        acc1 = __builtin_amdgcn_wmma_f32_16x16x4_f32(
            false, a2, false, b2, (short)0, acc1, false, false);
      }
      if (ncol < LQ) {
        const int ph = ncol / PW, pw = ncol - ph * PW;
#pragma unroll
        for (int v = 0; v < 8; ++v) {
          int   r   = mt * 16 + v + 8 * h;        // 0..431
          int   c   = r / 144, rem = r - c * 144;
          int   k1  = rem / KS, k2 = rem - k1 * KS;
          int   pix = (ph * KS + k1) * WID + pw * KS + k2;
          float val = acc0[v] + acc1[v];
          if (USE_WS) {
            // bijection (r, i<45) <-> (c, pix): each element written once
            part[(((size_t)b * NLAY + l) * 3 + c) * NPIX + pix] = val;
          } else {
            atomicAdd(&outacc[((size_t)b * 3 + c) * NPIX + pix], val);
          }
        }
      }
    }
  }
}

// ---------------------------------------------------------------------------
// ws path: sum 133 partial planes per output element + blend (one pass,
// coalesced: consecutive threads read consecutive pix within a plane).
// ---------------------------------------------------------------------------
__global__ void reduce_blend_kernel(const float* __restrict__ I,
                                    const float* __restrict__ M,
                                    const float* __restrict__ part,
                                    float* __restrict__ out, int n) {
  int i = blockIdx.x * blockDim.x + threadIdx.x;
  if (i < n) {
    int b   = i / (3 * NPIX);
    int rem = i - b * 3 * NPIX;     // c*NPIX + pix
    int pix = i % NPIX;
    const float* p = part + (size_t)b * NLAY * 3 * NPIX + rem;
    float s = 0.0f;
    for (int l = 0; l < NLAY; ++l) s += p[(size_t)l * 3 * NPIX];
    float m = M[(size_t)b * NPIX + pix];
    out[i] = I[i] * (1.0f - m) + s * m;
  }
}

// ---------------------------------------------------------------------------
// atomic-path helpers: d_out is poisoned by the harness -> zero it first.
// ---------------------------------------------------------------------------
__global__ void zero_kernel(float* __restrict__ out, int n) {
  int i = blockIdx.x * blockDim.x + threadIdx.x;
  if (i < n) out[i] = 0.0f;
}

__global__ void blend_kernel(const float* __restrict__ I,
                             const float* __restrict__ M,
                             float* __restrict__ out, int n) {
  int i = blockIdx.x * blockDim.x + threadIdx.x;
  if (i < n) {
    int b   = i / (3 * NPIX);
    int pix = i % NPIX;
    float m = M[(size_t)b * NPIX + pix];
    out[i] = I[i] * (1.0f - m) + out[i] * m;
  }
}

// ---------------------------------------------------------------------------
extern "C" void kernel_launch(void* const* d_in, const int* in_sizes, int n_in,
                              void* d_out, int out_size, void* d_ws, size_t ws_size,
                              hipStream_t stream) {
  const float* I = (const float*)d_in[0];
  const float* S = (const float*)d_in[1];
  const float* M = (const float*)d_in[2];
  float* out = (float*)d_out;

  const int    n    = NB * 3 * NPIX;                              // 77760
  const size_t need = (size_t)NB * NLAY * 3 * NPIX * sizeof(float); // ~39.5 MB

  if (d_ws != nullptr && ws_size >= need) {
    // streaming-partials path: 41 MB write + 41 MB read (~3.6 us @ 23.3 TB/s)
    float* part = (float*)d_ws;
    matting_layer_kernel<true><<<dim3(NLAY, NB), 288, 0, stream>>>(
        I, S, M, out, part);
    reduce_blend_kernel<<<(n + 255) / 256, 256, 0, stream>>>(I, M, part, out, n);
  } else {
    // fallback: accumulate with global atomics directly into d_out
    zero_kernel<<<(n + 255) / 256, 256, 0, stream>>>(out, n);
    matting_layer_kernel<false><<<dim3(NLAY, NB), 288, 0, stream>>>(
        I, S, M, out, nullptr);
    blend_kernel<<<(n + 255) / 256, 256, 0, stream>>>(I, M, out, n);
  }
}